// CascadeLayer_49323404427966
// MI455X (gfx1250) — compile-verified
//
#include <hip/hip_runtime.h>

typedef __attribute__((ext_vector_type(2))) float v2f;
typedef __attribute__((ext_vector_type(8))) float v8f;

#define N_NODES 50000
#define N_EDGES 800000
#define DIM     128
#define NALPHA  3

// ---------------------------------------------------------------------------
// H[m,n] = sum_k X[m,k] * W[n,k]   (FUSE: + bias[n], relu)
// One block = 8 waves = one 16-row tile; wave w handles output cols [16w,16w+16).
// Uses V_WMMA_F32_16X16X4_F32 (exact fp32) with K stepped by 4.
// ---------------------------------------------------------------------------
template <bool FUSE>
__global__ __launch_bounds__(256) void gemm_xwt_kernel(
    const float* __restrict__ X, const float* __restrict__ W,
    const float* __restrict__ bias, float* __restrict__ H)
{
    const int lane = threadIdx.x & 31;
    const int wave = threadIdx.x >> 5;        // 0..7 -> column tile
    const int m0   = blockIdx.x * 16;
    const int n0   = wave * 16;
    const int li   = lane & 15;               // M (for A) / N (for B,C,D)
    const int hi   = lane >> 4;               // K-pair select / row+8 select

    const float* __restrict__ xrow = X + (size_t)(m0 + li) * DIM + 2 * hi;
    const float* __restrict__ wrow = W + (size_t)(n0 + li) * DIM + 2 * hi;

    // Loop-invariant bias (single load, branch-free epilogue).
    const float bb = FUSE ? bias[n0 + li] : 0.0f;

    v8f c = {0.f, 0.f, 0.f, 0.f, 0.f, 0.f, 0.f, 0.f};

    #pragma unroll 8
    for (int k0 = 0; k0 < DIM; k0 += 4) {
        // A 16x4: lanes 0-15 hold K=k0,k0+1 ; lanes 16-31 hold K=k0+2,k0+3
        v2f a = *(const v2f*)(xrow + k0);
        // B 4x16: B[k,n] = W[n,k] -> same contiguous float2 from W row n
        v2f b = *(const v2f*)(wrow + k0);
        c = __builtin_amdgcn_wmma_f32_16x16x4_f32(
                /*neg_a=*/false, a, /*neg_b=*/false, b,
                /*c_mod=*/(short)0, c, /*reuse_a=*/false, /*reuse_b=*/false);
    }

    // C/D layout: VGPR r -> row m0 + r + 8*hi, col n0 + li.
    // Single base pointer; row stride DIM folds into immediate store offsets.
    float* __restrict__ hp = H + (size_t)(m0 + 8 * hi) * DIM + (n0 + li);
    #pragma unroll
    for (int r = 0; r < 8; ++r) {
        float v = c[r];
        if (FUSE) v = fmaxf(v + bb, 0.0f);
        hp[r * DIM] = v;
    }
}

// ---------------------------------------------------------------------------
__global__ void zero_f32_kernel(float* __restrict__ p, int n) {
    int i = blockIdx.x * blockDim.x + threadIdx.x;
    if (i < n) p[i] = 0.0f;
}

__global__ void deg_scatter_kernel(const long long* __restrict__ col,
                                   const float* __restrict__ w,
                                   float* __restrict__ deg, int E) {
    int e = blockIdx.x * blockDim.x + threadIdx.x;
    if (e < E) atomicAdd(&deg[(int)col[e]], w[e]);
}

__global__ void deg_to_dinv_kernel(float* __restrict__ deg, int n) {
    int i = blockIdx.x * blockDim.x + threadIdx.x;
    if (i < n) {
        float d = deg[i];
        deg[i] = (d > 0.0f) ? rsqrtf(d) : 0.0f;
    }
}

// One wave per edge: each lane gathers 4 contiguous floats of h[row],
// scales by norm, scatter-adds into acc[col]. h and acc live in L2 (192 MB).
__global__ __launch_bounds__(256) void edge_agg_kernel(
    const long long* __restrict__ row, const long long* __restrict__ col,
    const float* __restrict__ w, const float* __restrict__ dinv,
    const float* __restrict__ H, float* __restrict__ acc, int E)
{
    int t    = blockIdx.x * blockDim.x + threadIdx.x;
    int e    = t >> 5;
    int lane = t & 31;
    if (e >= E) return;

    const int r = (int)row[e];
    const int c = (int)col[e];
    const float nrm = dinv[r] * w[e] * dinv[c];

    const float4 hv = *(const float4*)(H + (size_t)r * DIM + lane * 4);
    float* __restrict__ ap = acc + (size_t)c * DIM + lane * 4;
    atomicAdd(ap + 0, hv.x * nrm);
    atomicAdd(ap + 1, hv.y * nrm);
    atomicAdd(ap + 2, hv.z * nrm);
    atomicAdd(ap + 3, hv.w * nrm);
}

__global__ void bias_relu_kernel(float* __restrict__ out,
                                 const float* __restrict__ bias, int total) {
    int i = blockIdx.x * blockDim.x + threadIdx.x;
    if (i < total) {
        int n = i & (DIM - 1);
        out[i] = fmaxf(out[i] + bias[n], 0.0f);
    }
}

// ---------------------------------------------------------------------------
extern "C" void kernel_launch(void* const* d_in, const int* in_sizes, int n_in,
                              void* d_out, int out_size, void* d_ws, size_t ws_size,
                              hipStream_t stream) {
    const float*     x          = (const float*)d_in[0];
    const long long* edge_index = (const long long*)d_in[1];  // int64 [ALPHA,2,E]
    const float*     edge_attr  = (const float*)d_in[2];      // [ALPHA,E]
    const float*     lin_w      = (const float*)d_in[3];      // [D,D]
    const float*     lin_b      = (const float*)d_in[4];      // [D]
    const float*     conv_w     = (const float*)d_in[5];      // [ALPHA,D,D]
    const float*     conv_b     = (const float*)d_in[6];      // [ALPHA,D]
    float*           out        = (float*)d_out;              // 4 x [N,D]

    float* h   = (float*)d_ws;                     // N*D floats (25.6 MB)
    float* deg = h + (size_t)N_NODES * DIM;        // N floats (doubles as dinv)

    const int ND = N_NODES * DIM;

    // hs[0] = relu(x @ lin_w.T + lin_b)
    gemm_xwt_kernel<true><<<N_NODES / 16, 256, 0, stream>>>(x, lin_w, lin_b, out);

    for (int i = 0; i < NALPHA; ++i) {
        const long long* rowp = edge_index + (size_t)i * 2 * N_EDGES;
        const long long* colp = rowp + N_EDGES;
        const float*     wp   = edge_attr + (size_t)i * N_EDGES;
        float*           acc  = out + (size_t)(i + 1) * ND;

        zero_f32_kernel<<<(N_NODES + 255) / 256, 256, 0, stream>>>(deg, N_NODES);
        deg_scatter_kernel<<<(N_EDGES + 255) / 256, 256, 0, stream>>>(colp, wp, deg, N_EDGES);
        deg_to_dinv_kernel<<<(N_NODES + 255) / 256, 256, 0, stream>>>(deg, N_NODES);

        gemm_xwt_kernel<false><<<N_NODES / 16, 256, 0, stream>>>(
            x, conv_w + (size_t)i * DIM * DIM, nullptr, h);

        zero_f32_kernel<<<(ND + 255) / 256, 256, 0, stream>>>(acc, ND);
        edge_agg_kernel<<<(N_EDGES * 32 + 255) / 256, 256, 0, stream>>>(
            rowp, colp, wp, deg, h, acc, N_EDGES);
        bias_relu_kernel<<<(ND + 255) / 256, 256, 0, stream>>>(
            acc, conv_b + (size_t)i * DIM, ND);
    }
}